// EnhancedHamiltonianEvolution_31954556682836
// MI455X (gfx1250) — compile-verified
//
#include <hip/hip_runtime.h>
#include <math.h>

typedef float v4f __attribute__((ext_vector_type(4)));

namespace {
constexpr int kQD   = 512;    // quaternion feature dim
constexpr int kDM   = 2048;   // model dim = 4 * kQD
constexpr int kTPB  = 128;    // 4 waves (wave32)
constexpr int kGrid = 2048;   // blocks; each handles nrows/kGrid rows

__global__ __launch_bounds__(kTPB) void quat_rot_kernel(
    const float* __restrict__ x,
    const float* __restrict__ q_left,
    const float* __restrict__ q_right,
    const float* __restrict__ gate,
    float* __restrict__ out,
    int nrows)
{
    // LDS layout: [0,2048) q_left | [2048,4096) q_right | [4096,4608) gate
    __shared__ __align__(16) float sh[2 * kDM + kQD];

    const int tid = threadIdx.x;

    // ---- CDNA5 async global->LDS staging of the parameter tables ----------
    // 18 KB total = 1152 x 16B chunks = 128 threads x 9 issues.
    {
        const unsigned lds_base = (unsigned)(size_t)(&sh[0]);
        const unsigned long long a_ql = (unsigned long long)(size_t)q_left;
        const unsigned long long a_qr = (unsigned long long)(size_t)q_right;
        const unsigned long long a_g  = (unsigned long long)(size_t)gate;
#pragma unroll
        for (int it = 0; it < 4; ++it) {                    // 8 KB of q_left
            const unsigned chunk = (unsigned)(it * kTPB + tid);   // 0..511
            const unsigned lds   = lds_base + chunk * 16u;
            const unsigned voff  = chunk * 16u;
            asm volatile("global_load_async_to_lds_b128 %0, %1, %2 offset:0"
                         :: "v"(lds), "v"(voff), "s"(a_ql) : "memory");
        }
#pragma unroll
        for (int it = 0; it < 4; ++it) {                    // 8 KB of q_right
            const unsigned chunk = (unsigned)(it * kTPB + tid);
            const unsigned lds   = lds_base + (unsigned)kDM * 4u + chunk * 16u;
            const unsigned voff  = chunk * 16u;
            asm volatile("global_load_async_to_lds_b128 %0, %1, %2 offset:0"
                         :: "v"(lds), "v"(voff), "s"(a_qr) : "memory");
        }
        {                                                   // 2 KB of gate
            const unsigned chunk = (unsigned)tid;                 // 0..127
            const unsigned lds   = lds_base + (unsigned)(2 * kDM) * 4u + chunk * 16u;
            const unsigned voff  = chunk * 16u;
            asm volatile("global_load_async_to_lds_b128 %0, %1, %2 offset:0"
                         :: "v"(lds), "v"(voff), "s"(a_g) : "memory");
        }
        asm volatile("s_wait_asynccnt 0x0" ::: "memory");   // this wave's asyncs done
    }
    __syncthreads();                                        // all waves' copies visible

    // ---- Build per-channel fused matrices M_c = g * L(ql^) * R(conj(qr^)) --
    const int c0 = tid * 4;   // this thread owns channels c0..c0+3 forever
    float M[4][16];
    {
        v4f qlv[4], qrv[4], gv;
#pragma unroll
        for (int i = 0; i < 4; ++i) {
            qlv[i] = *(const v4f*)(&sh[i * kQD + c0]);
            qrv[i] = *(const v4f*)(&sh[kDM + i * kQD + c0]);
        }
        gv = *(const v4f*)(&sh[2 * kDM + c0]);

#pragma unroll
        for (int j = 0; j < 4; ++j) {
            const float a0 = qlv[0][j], a1 = qlv[1][j], a2 = qlv[2][j], a3 = qlv[3][j];
            const float b0 = qrv[0][j], b1 = qrv[1][j], b2 = qrv[2][j], b3 = qrv[3][j];

            const float sl = a0*a0 + a1*a1 + a2*a2 + a3*a3 + 1e-8f;
            const float il = 1.0f / sqrtf(sl);
            const float lw = a0*il, lx = a1*il, ly = a2*il, lz = a3*il;

            const float sr = b0*b0 + b1*b1 + b2*b2 + b3*b3 + 1e-8f;
            const float ir = 1.0f / sqrtf(sr);
            // conj(qr^): negate vector part
            const float pw = b0*ir, px = -b1*ir, py = -b2*ir, pz = -b3*ir;

            const float g = gv[j];

            // left-multiplication matrix of ql^ and right-multiplication matrix of p
            const float L[4][4] = {{lw,-lx,-ly,-lz},
                                   {lx, lw,-lz, ly},
                                   {ly, lz, lw,-lx},
                                   {lz,-ly, lx, lw}};
            const float R[4][4] = {{pw,-px,-py,-pz},
                                   {px, pw, pz,-py},
                                   {py,-pz, pw, px},
                                   {pz, py,-px, pw}};
#pragma unroll
            for (int r = 0; r < 4; ++r)
#pragma unroll
                for (int k = 0; k < 4; ++k) {
                    float acc = L[r][0] * R[0][k];
                    acc += L[r][1] * R[1][k];
                    acc += L[r][2] * R[2][k];
                    acc += L[r][3] * R[3][k];
                    M[j][r * 4 + k] = g * acc;
                }
        }
    }

    // ---- Stream rows: fully-coalesced non-temporal B128 in/out --------------
    for (int row = blockIdx.x; row < nrows; row += gridDim.x) {
        const float* xp = x   + (size_t)row * kDM + c0;
        float*       op = out + (size_t)row * kDM + c0;

        v4f xin[4];
#pragma unroll
        for (int comp = 0; comp < 4; ++comp)
            xin[comp] = __builtin_nontemporal_load((const v4f*)(xp + comp * kQD));

        v4f o[4];
#pragma unroll
        for (int j = 0; j < 4; ++j) {
            const float xw = xin[0][j], xx = xin[1][j], xy = xin[2][j], xz = xin[3][j];
            const float* m = M[j];
            o[0][j] = m[ 0]*xw + m[ 1]*xx + m[ 2]*xy + m[ 3]*xz;
            o[1][j] = m[ 4]*xw + m[ 5]*xx + m[ 6]*xy + m[ 7]*xz;
            o[2][j] = m[ 8]*xw + m[ 9]*xx + m[10]*xy + m[11]*xz;
            o[3][j] = m[12]*xw + m[13]*xx + m[14]*xy + m[15]*xz;
        }

#pragma unroll
        for (int comp = 0; comp < 4; ++comp)
            __builtin_nontemporal_store(o[comp], (v4f*)(op + comp * kQD));
    }
}
}  // namespace

extern "C" void kernel_launch(void* const* d_in, const int* in_sizes, int n_in,
                              void* d_out, int out_size, void* d_ws, size_t ws_size,
                              hipStream_t stream) {
    const float* x    = (const float*)d_in[0];  // [B,T,2048] f32
    const float* ql   = (const float*)d_in[1];  // [4,512] f32
    const float* qr   = (const float*)d_in[2];  // [4,512] f32
    const float* gate = (const float*)d_in[3];  // [1,1,1,512] f32
    float* out = (float*)d_out;

    const int nrows = in_sizes[0] / kDM;        // B*T = 16384
    const int grid  = nrows < kGrid ? nrows : kGrid;
    hipLaunchKernelGGL(quat_rot_kernel, dim3(grid), dim3(kTPB), 0, stream,
                       x, ql, qr, gate, out, nrows);
}